// GNN_34359738978
// MI455X (gfx1250) — compile-verified
//
#include <hip/hip_runtime.h>

typedef float v2f __attribute__((ext_vector_type(2)));
typedef float v8f __attribute__((ext_vector_type(8)));

#define DH 256   // hidden dim (output cols of every GEMM)

// ---------------------------------------------------------------------------
// Degree / normalization kernels
// ---------------------------------------------------------------------------
__global__ void deg_init(float* __restrict__ deg, int n) {
    int i = blockIdx.x * blockDim.x + threadIdx.x;
    if (i < n) deg[i] = 1.0f;                       // self-loop
}

__global__ void deg_accum(const long long* __restrict__ dst,
                          float* __restrict__ deg, int ne) {
    int e = blockIdx.x * blockDim.x + threadIdx.x;
    if (e < ne) atomicAdd(&deg[(int)dst[e]], 1.0f);
}

__global__ void deg_finalize(float* __restrict__ deg, int n) {
    int i = blockIdx.x * blockDim.x + threadIdx.x;
    if (i < n) deg[i] = rsqrtf(deg[i]);             // deg -> dinv in place
}

// ---------------------------------------------------------------------------
// WMMA fp32 GEMM:  Y[M,256] = (RELU ? max(X,0) : X)[M,K] @ W[K,256]
// block = 128 threads = 4 waves; block computes 16 rows x 256 cols.
// wave w computes 4 adjacent 16x16 tiles at n0 = 64*w (A-frag reuse x4).
// Uses V_WMMA_F32_16X16X4_F32 (full fp32 precision, f32 accumulate).
// ---------------------------------------------------------------------------
template <bool RELU>
__global__ __launch_bounds__(128)
void gemm_wmma_f32(const float* __restrict__ X, const float* __restrict__ W,
                   float* __restrict__ Y, int M, int K) {
    const int lane = threadIdx.x & 31;
    const int wave = threadIdx.x >> 5;
    const int m0   = blockIdx.x << 4;
    const int n0   = wave << 6;          // 0,64,128,192
    const int r    = lane & 15;          // A row-in-tile / B+C col-in-tile
    const int kh   = lane >> 4;          // K-half select (ISA f32 A/B layout)

    int arow = m0 + r;                   // clamp keeps EXEC uniform (WMMA req)
    if (arow >= M) arow = M - 1;
    const float* xrow = X + (size_t)arow * K;

    v8f c0 = {}, c1 = {}, c2 = {}, c3 = {};

    for (int k0 = 0; k0 < K; k0 += 4) {
        const int ka = k0 + (kh << 1);   // this lane holds K = ka, ka+1
        v2f a = *reinterpret_cast<const v2f*>(xrow + ka);
        if (RELU) { a.x = fmaxf(a.x, 0.0f); a.y = fmaxf(a.y, 0.0f); }

        const float* w0 = W + (size_t)ka * DH + n0 + r;   // row ka
        // b.x = W[ka][n], b.y = W[ka+1][n]
        v2f b0, b1, b2, b3;
        b0.x = w0[0];   b0.y = w0[DH];
        b1.x = w0[16];  b1.y = w0[DH + 16];
        b2.x = w0[32];  b2.y = w0[DH + 32];
        b3.x = w0[48];  b3.y = w0[DH + 48];

        c0 = __builtin_amdgcn_wmma_f32_16x16x4_f32(false, a, false, b0, (short)0, c0, false, false);
        c1 = __builtin_amdgcn_wmma_f32_16x16x4_f32(false, a, false, b1, (short)0, c1, false, false);
        c2 = __builtin_amdgcn_wmma_f32_16x16x4_f32(false, a, false, b2, (short)0, c2, false, false);
        c3 = __builtin_amdgcn_wmma_f32_16x16x4_f32(false, a, false, b3, (short)0, c3, false, false);
    }

    // C/D layout: vgpr i, lanes 0-15 -> M=i, lanes 16-31 -> M=8+i; N = lane&15
    const int n = n0 + r;
#pragma unroll
    for (int i = 0; i < 8; ++i) {
        const int m = m0 + i + (kh << 3);
        if (m < M) {
            float* yrow = Y + (size_t)m * DH + n;
            yrow[0]  = c0[i];
            yrow[16] = c1[i];
            yrow[32] = c2[i];
            yrow[48] = c3[i];
        }
    }
}

// ---------------------------------------------------------------------------
// out[i,f] = h[i,f] * dinv[i]^2 + b[f]   (self-loop + bias; also zero-inits
// the accumulator deterministically every call)
// ---------------------------------------------------------------------------
__global__ void init_agg(const float* __restrict__ h, const float* __restrict__ dinv,
                         const float* __restrict__ b, float* __restrict__ out,
                         long long total) {
    long long i = (long long)blockIdx.x * blockDim.x + threadIdx.x;
    if (i >= total) return;
    int node = (int)(i >> 8);
    int f    = (int)(i & 255);
    float d  = dinv[node];
    out[i] = h[i] * d * d + b[f];
}

// ---------------------------------------------------------------------------
// Edge scatter: one wave per edge; lane l handles features l + 32j.
// Coalesced gather of h[src], consecutive-lane f32 atomics into out[dst]
// (dst rows are L2-resident: 51 MB << 192 MB L2).
// ---------------------------------------------------------------------------
__global__ __launch_bounds__(256)
void edge_scatter(const float* __restrict__ h, const float* __restrict__ dinv,
                  const long long* __restrict__ src, const long long* __restrict__ dst,
                  float* __restrict__ out, int ne) {
    int e = blockIdx.x * 8 + (threadIdx.x >> 5);
    if (e >= ne) return;
    const int lane = threadIdx.x & 31;
    const int s = (int)src[e];
    const int d = (int)dst[e];
    const float coef = dinv[s] * dinv[d];
    const float* hs = h + (size_t)s * DH;
    float*       od = out + (size_t)d * DH;
#pragma unroll
    for (int j = 0; j < 8; ++j) {
        const int f = lane + (j << 5);
        atomicAdd(&od[f], hs[f] * coef);
    }
}

// ---------------------------------------------------------------------------
extern "C" void kernel_launch(void* const* d_in, const int* in_sizes, int n_in,
                              void* d_out, int out_size, void* d_ws, size_t ws_size,
                              hipStream_t stream) {
    const float*     x   = (const float*)d_in[0];
    const long long* ei  = (const long long*)d_in[1];   // int64 per reference
    const float*     W1  = (const float*)d_in[2];
    const float*     b1  = (const float*)d_in[3];
    const float*     W2  = (const float*)d_in[4];
    const float*     b2  = (const float*)d_in[5];
    const float*     W3  = (const float*)d_in[6];
    const float*     b3  = (const float*)d_in[7];
    float*           out = (float*)d_out;

    const int D_IN    = 128;
    const int n_nodes = in_sizes[0] / D_IN;
    const int n_edges = in_sizes[1] / 2;
    const long long* src = ei;
    const long long* dst = ei + n_edges;

    // workspace: dinv | bufA (51.2 MB) | bufB (51.2 MB)
    char*  ws   = (char*)d_ws;
    float* dinv = (float*)ws;
    size_t featBytes = (size_t)n_nodes * DH * sizeof(float);
    float* bufA = (float*)(ws + (1u << 18));
    float* bufB = (float*)(ws + (1u << 18) + featBytes);

    const int TB = 256;
    const long long total = (long long)n_nodes * DH;
    const int nodeBlks = (n_nodes + TB - 1) / TB;
    const int edgeBlks = (n_edges + TB - 1) / TB;
    const int aggBlks  = (int)((total + TB - 1) / TB);
    const int scatBlks = (n_edges + 7) / 8;
    const int mtiles   = (n_nodes + 15) / 16;

    // --- normalization coefficients ---
    deg_init    <<<nodeBlks, TB, 0, stream>>>(dinv, n_nodes);
    deg_accum   <<<edgeBlks, TB, 0, stream>>>(dst, dinv, n_edges);
    deg_finalize<<<nodeBlks, TB, 0, stream>>>(dinv, n_nodes);

    // --- layer 1: h1 = x @ W1 ; out1 = scatter(h1) + selfloop + b1 ---
    gemm_wmma_f32<false><<<mtiles, 128, 0, stream>>>(x, W1, bufA, n_nodes, D_IN);
    init_agg    <<<aggBlks, TB, 0, stream>>>(bufA, dinv, b1, bufB, total);
    edge_scatter<<<scatBlks, TB, 0, stream>>>(bufA, dinv, src, dst, bufB, n_edges);

    // --- layer 2 (ReLU of layer-1 result fused into GEMM A-load) ---
    gemm_wmma_f32<true><<<mtiles, 128, 0, stream>>>(bufB, W2, bufA, n_nodes, DH);
    init_agg    <<<aggBlks, TB, 0, stream>>>(bufA, dinv, b2, bufB, total);
    edge_scatter<<<scatBlks, TB, 0, stream>>>(bufA, dinv, src, dst, bufB, n_edges);

    // --- layer 3 (ReLU fused; no ReLU on final output) ---
    gemm_wmma_f32<true><<<mtiles, 128, 0, stream>>>(bufB, W3, bufA, n_nodes, DH);
    init_agg    <<<aggBlks, TB, 0, stream>>>(bufA, dinv, b3, out, total);
    edge_scatter<<<scatBlks, TB, 0, stream>>>(bufA, dinv, src, dst, out, n_edges);
}